// Attn_82343112999416
// MI455X (gfx1250) — compile-verified
//
#include <hip/hip_runtime.h>

// Shapes from the reference
#define HID 4096
#define NOBJ 8192
#define RBLK 8   // K-reduction split per GEMV

typedef float v2f __attribute__((ext_vector_type(2)));
typedef float v8f __attribute__((ext_vector_type(8)));

// Computes partial[rblk][j] = sum_{i in row-block} vec[i] * mat[i][j]
// using V_WMMA_F32_16X16X4_F32 with A = vec segment replicated over all 16 rows.
// Each wave owns one 16-column tile of one row-block. COLS / ROWS_PER_BLK are
// compile-time so all strides fold into immediate load offsets.
template <int COLS, int ROWS_PER_BLK>
__global__ __launch_bounds__(128)
void matvec_wmma_partial(const float* __restrict__ vec,
                         const float* __restrict__ mat,
                         float* __restrict__ part)
{
    constexpr int TILES = COLS / 16;

    const int lane = threadIdx.x & 31;
    const int wave = threadIdx.x >> 5;
    const int w    = blockIdx.x * 4 + wave;
    const int tile = w % TILES;
    const int rblk = w / TILES;
    const int j0   = tile << 4;
    const int i0   = rblk * ROWS_PER_BLK;

    const int hi = lane >> 4;   // 0: K=0,1   1: K=2,3   (A/B 32-bit layouts)
    const int lo = lane & 15;   // column within tile / M row

    v8f acc = {0.f, 0.f, 0.f, 0.f, 0.f, 0.f, 0.f, 0.f};

    const float* mrow = mat + (size_t)(i0 + 2 * hi) * COLS + j0 + lo;
    const float* vp   = vec + i0 + 2 * hi;

    #pragma unroll 16
    for (int k = 0; k < ROWS_PER_BLK; k += 4) {
        v2f a, b;
        // A (16x4, replicated rows): VGPR0 = A[*][K=2*hi], VGPR1 = A[*][K=2*hi+1]
        a.x = vp[0];
        a.y = vp[1];
        // B (4x16): VGPR0 = B[K=2*hi][lo], VGPR1 = B[K=2*hi+1][lo]
        b.x = mrow[0];
        b.y = mrow[COLS];
        acc = __builtin_amdgcn_wmma_f32_16x16x4_f32(
            /*neg_a=*/false, a, /*neg_b=*/false, b,
            /*c_mod=*/(short)0, acc, /*reuse_a=*/false, /*reuse_b=*/false);
        vp   += 4;
        mrow += (size_t)4 * COLS;
    }

    // Every row of D equals the GEMV partial; row M=0 lives in acc[0], lanes 0..15.
    if (hi == 0) {
        part[(size_t)rblk * COLS + j0 + lo] = acc[0];
    }
}

__global__ void reduce_partials(const float* __restrict__ part,
                                float* __restrict__ outv, int cols)
{
    int j = blockIdx.x * blockDim.x + threadIdx.x;
    if (j < cols) {
        float s = 0.f;
        #pragma unroll
        for (int r = 0; r < RBLK; ++r) s += part[(size_t)r * cols + j];
        outv[j] = s;
    }
}

// Reduce the 8 energy partials and softmax over all 8192 objects in one block.
__global__ __launch_bounds__(1024)
void softmax_energies(const float* __restrict__ partB, float* __restrict__ out)
{
    __shared__ float red[1024];
    const int t = threadIdx.x;

    float e[8];
    float m = -__builtin_inff();
    #pragma unroll
    for (int k = 0; k < 8; ++k) {
        int n = t + k * 1024;
        float s = 0.f;
        #pragma unroll
        for (int r = 0; r < RBLK; ++r) s += partB[(size_t)r * NOBJ + n];
        e[k] = s;
        m = fmaxf(m, s);
    }

    red[t] = m;
    __syncthreads();
    for (int s = 512; s > 0; s >>= 1) {
        if (t < s) red[t] = fmaxf(red[t], red[t + s]);
        __syncthreads();
    }
    m = red[0];
    __syncthreads();

    float sum = 0.f;
    #pragma unroll
    for (int k = 0; k < 8; ++k) { e[k] = __expf(e[k] - m); sum += e[k]; }

    red[t] = sum;
    __syncthreads();
    for (int s = 512; s > 0; s >>= 1) {
        if (t < s) red[t] += red[t + s];
        __syncthreads();
    }
    float inv = 1.0f / red[0];

    #pragma unroll
    for (int k = 0; k < 8; ++k) out[t + k * 1024] = e[k] * inv;
}

extern "C" void kernel_launch(void* const* d_in, const int* in_sizes, int n_in,
                              void* d_out, int out_size, void* d_ws, size_t ws_size,
                              hipStream_t stream)
{
    const float* hidden = (const float*)d_in[0];   // [H]
    const float* objs   = (const float*)d_in[1];   // [H, N]
    const float* W      = (const float*)d_in[2];   // [H, H]
    // d_in[3] (b) intentionally unused: hidden.b is constant over objects and
    // softmax is shift-invariant, so it cancels exactly.
    float* out = (float*)d_out;                    // [1, N] = 8192 floats

    float* partA = (float*)d_ws;                   // RBLK * H floats
    float* v     = partA + (size_t)RBLK * HID;     // H floats
    float* partB = v + HID;                        // RBLK * N floats
    (void)in_sizes; (void)n_in; (void)out_size; (void)ws_size;

    // Phase 1: v = W^T * hidden.  256 col-tiles x 8 row-blocks = 2048 waves.
    matvec_wmma_partial<HID, HID / RBLK><<<512, 128, 0, stream>>>(hidden, W, partA);
    reduce_partials<<<HID / 256, 256, 0, stream>>>(partA, v, HID);

    // Phase 2: e = objs^T * v.  512 col-tiles x 8 row-blocks = 4096 waves.
    matvec_wmma_partial<NOBJ, HID / RBLK><<<1024, 128, 0, stream>>>(v, objs, partB);

    // Phase 3: softmax over the 8192 energies.
    softmax_energies<<<1, 1024, 0, stream>>>(partB, out);
}